// DecoderStepWrapper_1709396983904
// MI455X (gfx1250) — compile-verified
//
#include <hip/hip_runtime.h>
#include <hip/hip_bf16.h>
#include <float.h>

// Model dims (fixed by the reference)
#define LN   16
#define DIM  1024
#define NH   16
#define NKV  8
#define DH   128
#define SEQ  4096
#define FF   4096
#define VOC  32000
#define REP  (NH / NKV)   // 2
#define KT   32           // k-rows per TDM weight tile

typedef __attribute__((ext_vector_type(2))) float v2f;
typedef __attribute__((ext_vector_type(8))) float v8f;
typedef __attribute__((ext_vector_type(4))) unsigned int v4u;
typedef __attribute__((ext_vector_type(4))) int v4i;
typedef __attribute__((ext_vector_type(8))) int v8i;

// ---------------------------------------------------------------------------
// TDM: issue one tensor_load_to_lds of a [rows x cols] fp32 tile from a
// row-major tensor with row stride `n` into LDS at byte offset lds_off.
// Descriptor layout per CDNA5 ISA §8.3/§8.4 (D# group0/group1).
// This toolchain exposes the 6-arg builtin (g0, g1, g2, g3, g_extra, cpol).
// ---------------------------------------------------------------------------
__device__ __forceinline__ void tdm_load_tile(unsigned lds_off,
                                              const float* gptr,
                                              unsigned rows, unsigned cols,
                                              unsigned n, unsigned tensor_rows) {
  const unsigned long long ga = (unsigned long long)(uintptr_t)gptr;
  v4u g0;
  g0[0] = 1u;                                        // count=1 (valid user D#)
  g0[1] = lds_off;                                   // lds_addr (bytes)
  g0[2] = (unsigned)(ga & 0xffffffffu);              // global_addr[31:0]
  g0[3] = (unsigned)((ga >> 32) & 0x1ffffffu)        // global_addr[56:32]
        | (2u << 30);                                // type = 2 ("image")
  v8i g1;
  g1[0] = (int)(2u << 16);                           // data_size=2 (4B), mask=0
  g1[1] = (int)((n & 0xffffu) << 16);                // tensor_dim0[15:0] @63:48
  g1[2] = (int)(((n >> 16) & 0xffffu)                // tensor_dim0[31:16]
        | ((tensor_rows & 0xffffu) << 16));          // tensor_dim1[15:0]
  g1[3] = (int)(((tensor_rows >> 16) & 0xffffu)      // tensor_dim1[31:16]
        | ((cols & 0xffffu) << 16));                 // tile_dim0 @127:112
  g1[4] = (int)(rows & 0xffffu);                     // tile_dim1 (tile_dim2=0)
  g1[5] = (int)n;                                    // tensor_dim0_stride[31:0]
  g1[6] = 0;                                         // stride[47:32], dim1_stride lo
  g1[7] = 0;
  const v4i z4 = {0, 0, 0, 0};
  const v8i z8 = {0, 0, 0, 0, 0, 0, 0, 0};
  __builtin_amdgcn_tensor_load_to_lds(g0, g1, z4, z4, z8, 0);
}

// ---------------------------------------------------------------------------
// Embedding gather: x = embed[token]
// ---------------------------------------------------------------------------
__global__ void embed_kernel(const float* __restrict__ embed,
                             const int* __restrict__ input_ids,
                             float* __restrict__ x) {
  const int tok = input_ids[0];
  const int i = blockIdx.x * blockDim.x + threadIdx.x;
  if (i < DIM) x[i] = embed[(size_t)tok * DIM + i];
}

// ---------------------------------------------------------------------------
// RMSNorm: out = x * rsqrt(mean(x^2)+eps) * w   (one block)
// ---------------------------------------------------------------------------
__global__ void rmsnorm_kernel(const float* __restrict__ x,
                               const float* __restrict__ w,
                               float* __restrict__ out, int d) {
  __shared__ float red[256];
  const int tid = threadIdx.x;
  float ss = 0.f;
  for (int i = tid; i < d; i += 256) { float v = x[i]; ss += v * v; }
  red[tid] = ss; __syncthreads();
  for (int off = 128; off; off >>= 1) {
    if (tid < off) red[tid] += red[tid + off];
    __syncthreads();
  }
  const float r = rsqrtf(red[0] / (float)d + 1e-6f);
  for (int i = tid; i < d; i += 256) out[i] = x[i] * r * w[i];
}

// ---------------------------------------------------------------------------
// WMMA GEMV with TDM double-buffered weight staging.
//   y[col] (+)= sum_k x[k] * W[k*N + col]
// Block = 256 threads (8 waves) -> 128-column tile.
// Wave 0 drives the Tensor Data Mover: [KT x 128] fp32 weight tiles are DMAed
// into LDS (tensor_load_to_lds, TENSORcnt), ping-pong buffered; all 8 waves
// run V_WMMA_F32_16X16X4_F32 with the B fragment read from LDS.
//   A (16x4): row 0 = x[k..k+3] (lane 0: K=0,1; lane 16: K=2,3), rest zero.
//   B (4x16): vgpr0 = rows {k, k+2}, vgpr1 = rows {k+1, k+3}, lanes = cols.
//   D row 0 (VGPR0, lanes 0..15) = the 16 outputs.
// ---------------------------------------------------------------------------
__global__ void gemv_wmma_kernel(const float* __restrict__ x,
                                 const float* __restrict__ W,
                                 float* __restrict__ y,
                                 int d, int n, int accumulate) {
  __shared__ float xs[4096];
  __shared__ float wb[2][KT * 128];

  const int tid  = threadIdx.x;
  const int lane = tid & 31;
  const int wave = tid >> 5;
  const int colblock = blockIdx.x * 128;

  for (int i = tid; i < d; i += 256) xs[i] = x[i];

  const int colLocal = wave * 16 + (lane & 15);
  const int half = lane >> 4;                   // 0: lanes 0-15, 1: 16-31
  const int ntiles = d / KT;

  // Prologue: DMA tile 0 into buffer 0.
  if (wave == 0) {
    tdm_load_tile((unsigned)(uintptr_t)(void*)&wb[0][0],
                  W + colblock, KT, 128, (unsigned)n, (unsigned)d);
    __builtin_amdgcn_s_wait_tensorcnt(0);
  }
  __syncthreads();

  v8f acc = {};
  for (int t = 0; t < ntiles; ++t) {
    const int cur = t & 1;
    // Issue DMA for the next tile while computing on the current one.
    if (wave == 0 && (t + 1) < ntiles) {
      tdm_load_tile((unsigned)(uintptr_t)(void*)&wb[cur ^ 1][0],
                    W + (size_t)(t + 1) * KT * n + colblock,
                    KT, 128, (unsigned)n, (unsigned)d);
    }

    const float* wt = &wb[cur][0];
    const int kbase = t * KT;
    #pragma unroll
    for (int kk = 0; kk < KT; kk += 4) {
      // A fragment: activation chunk in matrix row 0 only
      v2f a; a.x = 0.f; a.y = 0.f;
      if ((lane & 15) == 0) {
        const int g = kbase + kk + (half << 1);
        a.x = xs[g]; a.y = xs[g + 1];
      }
      // B fragment from the LDS-staged tile
      v2f b;
      b.x = wt[(kk + 2 * half)     * 128 + colLocal];
      b.y = wt[(kk + 2 * half + 1) * 128 + colLocal];
      acc = __builtin_amdgcn_wmma_f32_16x16x4_f32(
          /*neg_a=*/false, a, /*neg_b=*/false, b,
          /*c_mod=*/(short)0, acc, /*reuse_a=*/false, /*reuse_b=*/false);
    }

    if (wave == 0 && (t + 1) < ntiles) __builtin_amdgcn_s_wait_tensorcnt(0);
    __syncthreads();
  }

  if (lane < 16) {                              // D row 0 = VGPR0, lanes 0..15
    const int col = colblock + wave * 16 + lane;
    const float r = acc[0];
    if (col < n) {
      if (accumulate) y[col] += r;
      else            y[col]  = r;
    }
  }
}

// ---------------------------------------------------------------------------
// RoPE on q (in place) and k, and write k/v into the output KV cache at pos.
// blocks 0..15: q heads; 16..23: k heads (rope + cache write); 24..31: v write
// ---------------------------------------------------------------------------
__global__ void rope_cache_kernel(float* __restrict__ q,
                                  const float* __restrict__ kbuf,
                                  const float* __restrict__ vbuf,
                                  float* __restrict__ kcache,  // [NKV,SEQ,DH]
                                  float* __restrict__ vcache,
                                  const int* __restrict__ prefill,
                                  const int* __restrict__ step) {
  const int b = blockIdx.x, t = threadIdx.x;     // 128 threads
  const int pos = prefill[0] + step[0];
  const float fpos = (float)pos;
  const int i = (t < 64) ? t : t - 64;
  const float ang = fpos * __expf(-((float)i / 64.f) * __logf(10000.f));
  const float c = __cosf(ang), s = __sinf(ang);

  if (b < NH) {                                   // rope q in place
    __shared__ float sh[DH];
    float* qh = q + (size_t)b * DH;
    sh[t] = qh[t]; __syncthreads();
    qh[t] = (t < 64) ? (sh[t] * c - sh[t + 64] * s)
                     : (sh[t] * c + sh[t - 64] * s);
  } else if (b < NH + NKV) {                      // rope k -> cache
    const int j = b - NH;
    const float* src = kbuf + (size_t)j * DH;
    float* dst = kcache + ((size_t)j * SEQ + pos) * DH;
    dst[t] = (t < 64) ? (src[t] * c - src[t + 64] * s)
                      : (src[t] * c + src[t - 64] * s);
  } else {                                        // copy v -> cache
    const int j = b - NH - NKV;
    vcache[((size_t)j * SEQ + pos) * DH + t] = vbuf[(size_t)j * DH + t];
  }
}

// ---------------------------------------------------------------------------
// Attention scores: one wave per (head, seq row). grid = (SEQ/8, NH)
// ---------------------------------------------------------------------------
__global__ void scores_kernel(const float* __restrict__ q,
                              const float* __restrict__ kcache,
                              float* __restrict__ sc,
                              const int* __restrict__ prefill,
                              const int* __restrict__ step) {
  const int h = blockIdx.y;
  const int g = h >> 1;                            // GQA: rep = 2
  __shared__ float qs[DH];
  const int tid = threadIdx.x;
  if (tid < DH) qs[tid] = q[(size_t)h * DH + tid];
  __syncthreads();

  const int pos  = prefill[0] + step[0];
  const int lane = tid & 31, wave = tid >> 5;
  const int s = blockIdx.x * 8 + wave;
  const float* kr = kcache + ((size_t)g * SEQ + s) * DH;
  float d = 0.f;
  #pragma unroll
  for (int i = lane; i < DH; i += 32) d += qs[i] * kr[i];
  for (int off = 16; off; off >>= 1) d += __shfl_down(d, off, 32);
  if (lane == 0) {
    const float scale = 0.08838834764831845f;      // 1/sqrt(128)
    sc[(size_t)h * SEQ + s] = (s <= pos) ? d * scale : -3.402823466e38f;
  }
}

// Softmax over SEQ per head (one block per head), in place.
__global__ void softmax_kernel(float* __restrict__ sc) {
  const int h = blockIdx.x, tid = threadIdx.x;
  float* p = sc + (size_t)h * SEQ;
  __shared__ float red[256];
  float m = -3.402823466e38f;
  for (int i = tid; i < SEQ; i += 256) m = fmaxf(m, p[i]);
  red[tid] = m; __syncthreads();
  for (int off = 128; off; off >>= 1) {
    if (tid < off) red[tid] = fmaxf(red[tid], red[tid + off]);
    __syncthreads();
  }
  m = red[0]; __syncthreads();
  float sum = 0.f;
  for (int i = tid; i < SEQ; i += 256) sum += __expf(p[i] - m);
  red[tid] = sum; __syncthreads();
  for (int off = 128; off; off >>= 1) {
    if (tid < off) red[tid] += red[tid + off];
    __syncthreads();
  }
  const float inv = 1.f / red[0];
  for (int i = tid; i < SEQ; i += 256) p[i] = __expf(p[i] - m) * inv;
}

// attn[h,d] = sum_s p[h,s] * V[g,s,d]   (one block per head, 128 threads)
__global__ void pv_kernel(const float* __restrict__ sc,
                          const float* __restrict__ vcache,
                          float* __restrict__ attn) {
  const int h = blockIdx.x, d = threadIdx.x;
  const int g = h >> 1;
  const float* p = sc + (size_t)h * SEQ;
  const float* vb = vcache + (size_t)g * SEQ * DH + d;
  float acc = 0.f;
  for (int s = 0; s < SEQ; ++s) {
    if ((s & 63) == 0 && s + 64 < SEQ)
      __builtin_prefetch(&vb[(size_t)(s + 64) * DH], 0, 1);
    acc += p[s] * vb[(size_t)s * DH];
  }
  attn[(size_t)h * DH + d] = acc;
}

// SwiGLU: h = silu(g) * u
__global__ void swiglu_kernel(const float* __restrict__ g,
                              const float* __restrict__ u,
                              float* __restrict__ h, int n) {
  const int i = blockIdx.x * blockDim.x + threadIdx.x;
  if (i < n) {
    const float gv = g[i];
    h[i] = (gv / (1.f + __expf(-gv))) * u[i];
  }
}

// ---------------------------------------------------------------------------
// Deterministic 2-stage argmax (first index on ties)
// ---------------------------------------------------------------------------
__global__ void argmax1_kernel(const float* __restrict__ v, int n,
                               float* __restrict__ bv, int* __restrict__ bi) {
  __shared__ float sv[256]; __shared__ int si[256];
  const int tid = threadIdx.x;
  float best = -3.402823466e38f; int bidx = 0x7fffffff;
  for (int i = blockIdx.x * 256 + tid; i < n; i += gridDim.x * 256) {
    const float x = v[i];
    if (x > best || (x == best && i < bidx)) { best = x; bidx = i; }
  }
  sv[tid] = best; si[tid] = bidx; __syncthreads();
  for (int off = 128; off; off >>= 1) {
    if (tid < off) {
      if (sv[tid + off] > sv[tid] ||
          (sv[tid + off] == sv[tid] && si[tid + off] < si[tid])) {
        sv[tid] = sv[tid + off]; si[tid] = si[tid + off];
      }
    }
    __syncthreads();
  }
  if (tid == 0) { bv[blockIdx.x] = sv[0]; bi[blockIdx.x] = si[0]; }
}

__global__ void argmax2_kernel(const float* __restrict__ bv,
                               const int* __restrict__ bi, int n,
                               float* __restrict__ out) {
  __shared__ float sv[64]; __shared__ int si[64];
  const int tid = threadIdx.x;
  sv[tid] = (tid < n) ? bv[tid] : -3.402823466e38f;
  si[tid] = (tid < n) ? bi[tid] : 0x7fffffff;
  __syncthreads();
  for (int off = 32; off; off >>= 1) {
    if (tid < off) {
      if (sv[tid + off] > sv[tid] ||
          (sv[tid + off] == sv[tid] && si[tid + off] < si[tid])) {
        sv[tid] = sv[tid + off]; si[tid] = si[tid + off];
      }
    }
    __syncthreads();
  }
  if (tid == 0) out[0] = (float)si[0];
}

// ---------------------------------------------------------------------------
// Driver
// ---------------------------------------------------------------------------
extern "C" void kernel_launch(void* const* d_in, const int* in_sizes, int n_in,
                              void* d_out, int out_size, void* d_ws, size_t ws_size,
                              hipStream_t stream) {
  (void)in_sizes; (void)n_in; (void)out_size; (void)ws_size;
  const int*   input_ids = (const int*)  d_in[0];
  const int*   step      = (const int*)  d_in[1];
  const int*   prefill   = (const int*)  d_in[2];
  const float* past_k    = (const float*)d_in[3];
  const float* past_v    = (const float*)d_in[4];
  const float* embed     = (const float*)d_in[5];
  const float* Wq  = (const float*)d_in[6];
  const float* Wk  = (const float*)d_in[7];
  const float* Wv  = (const float*)d_in[8];
  const float* Wo  = (const float*)d_in[9];
  const float* Wg  = (const float*)d_in[10];
  const float* Wu  = (const float*)d_in[11];
  const float* Wd  = (const float*)d_in[12];
  const float* ln1 = (const float*)d_in[13];
  const float* ln2 = (const float*)d_in[14];
  const float* lnf = (const float*)d_in[15];
  const float* Wlm = (const float*)d_in[16];

  const size_t kv_elems = (size_t)LN * NKV * SEQ * DH;   // 67,108,864
  float* out   = (float*)d_out;
  float* out_k = out + 1;            // new_keys  [L,1,KV,S,Dh]
  float* out_v = out_k + kv_elems;   // new_values

  // Copy caches wholesale; per-layer kernels overwrite row `pos` before use.
  (void)hipMemcpyAsync(out_k, past_k, kv_elems * sizeof(float),
                       hipMemcpyDeviceToDevice, stream);
  (void)hipMemcpyAsync(out_v, past_v, kv_elems * sizeof(float),
                       hipMemcpyDeviceToDevice, stream);

  // Workspace layout (floats)
  float* ws     = (float*)d_ws;
  float* x      = ws;                 size_t o = DIM;       // 1024
  float* nbuf   = ws + o;             o += DIM;             // 1024
  float* q      = ws + o;             o += NH * DH;         // 2048
  float* kbuf   = ws + o;             o += NKV * DH;        // 1024
  float* vbuf   = ws + o;             o += NKV * DH;        // 1024
  float* attn   = ws + o;             o += NH * DH;         // 2048
  float* sc     = ws + o;             o += (size_t)NH * SEQ;// 65536
  float* gbuf   = ws + o;             o += FF;              // 4096
  float* ubuf   = ws + o;             o += FF;              // 4096
  float* hbuf   = ws + o;             o += FF;              // 4096
  float* logits = ws + o;             o += VOC;             // 32000
  float* bv     = ws + o;             o += 64;
  int*   bi     = (int*)(ws + o);     o += 64;

  embed_kernel<<<4, 256, 0, stream>>>(embed, input_ids, x);

  for (int l = 0; l < LN; ++l) {
    float* kc = out_k + (size_t)l * NKV * SEQ * DH;
    float* vc = out_v + (size_t)l * NKV * SEQ * DH;

    // ---- attention ----
    rmsnorm_kernel<<<1, 256, 0, stream>>>(x, ln1 + (size_t)l * DIM, nbuf, DIM);
    gemv_wmma_kernel<<<(NH * DH) / 128, 256, 0, stream>>>(
        nbuf, Wq + (size_t)l * DIM * NH * DH, q, DIM, NH * DH, 0);
    gemv_wmma_kernel<<<(NKV * DH) / 128, 256, 0, stream>>>(
        nbuf, Wk + (size_t)l * DIM * NKV * DH, kbuf, DIM, NKV * DH, 0);
    gemv_wmma_kernel<<<(NKV * DH) / 128, 256, 0, stream>>>(
        nbuf, Wv + (size_t)l * DIM * NKV * DH, vbuf, DIM, NKV * DH, 0);
    rope_cache_kernel<<<NH + 2 * NKV, 128, 0, stream>>>(
        q, kbuf, vbuf, kc, vc, prefill, step);
    scores_kernel<<<dim3(SEQ / 8, NH), 256, 0, stream>>>(q, kc, sc, prefill, step);
    softmax_kernel<<<NH, 256, 0, stream>>>(sc);
    pv_kernel<<<NH, DH, 0, stream>>>(sc, vc, attn);
    gemv_wmma_kernel<<<DIM / 128, 256, 0, stream>>>(
        attn, Wo + (size_t)l * NH * DH * DIM, x, NH * DH, DIM, 1); // residual

    // ---- SwiGLU MLP ----
    rmsnorm_kernel<<<1, 256, 0, stream>>>(x, ln2 + (size_t)l * DIM, nbuf, DIM);
    gemv_wmma_kernel<<<FF / 128, 256, 0, stream>>>(
        nbuf, Wg + (size_t)l * DIM * FF, gbuf, DIM, FF, 0);
    gemv_wmma_kernel<<<FF / 128, 256, 0, stream>>>(
        nbuf, Wu + (size_t)l * DIM * FF, ubuf, DIM, FF, 0);
    swiglu_kernel<<<FF / 256, 256, 0, stream>>>(gbuf, ubuf, hbuf, FF);
    gemv_wmma_kernel<<<DIM / 128, 256, 0, stream>>>(
        hbuf, Wd + (size_t)l * FF * DIM, x, FF, DIM, 1);           // residual
  }

  // ---- LM head + argmax ----
  rmsnorm_kernel<<<1, 256, 0, stream>>>(x, lnf, nbuf, DIM);
  gemv_wmma_kernel<<<VOC / 128, 256, 0, stream>>>(nbuf, Wlm, logits, DIM, VOC, 0);
  argmax1_kernel<<<64, 256, 0, stream>>>(logits, VOC, bv, bi);
  argmax2_kernel<<<1, 64, 0, stream>>>(bv, bi, 64, out);
}